// LinearClassifyLayer_22144851378521
// MI455X (gfx1250) — compile-verified
//
#include <hip/hip_runtime.h>
#include <hip/hip_bf16.h>

// ---------------------------------------------------------------------------
// SNN forward: i = x@W^T (fp32 WMMA GEMM with async-to-LDS double-buffered
// staging), then T=500-step LIF scan with Hebbian plasticity. mem (64MB)
// stays L2-resident; rank-1 update fused lazily into next step's matvec pass
// (2 L2 passes/step instead of 3).
// ---------------------------------------------------------------------------

typedef __attribute__((ext_vector_type(2))) float v2f;
typedef __attribute__((ext_vector_type(8))) float v8f;

#define Bdim 64
#define Tdim 500
#define Idim 2048
#define Hdim 2048
#define Odim 128

#define ALPHA_C  0.90483741803595952f   /* exp(-1/10) */
#define DECAY_C  0.95122942450071400f   /* exp(-1/20) */
#define ONEMD_C  0.04877057549928600f   /* 1 - exp(-1/20) */
#define LR_C     1.0e-4f
#define THR_C    1.0f

// ------------------------------ GEMM ---------------------------------------
// C[M,N] = A[M,K] * B[N,K]^T with fp32 WMMA 16x16x4.
// Block tile 64x64, BK=16, 8 waves; each wave computes a 16x32 strip.
// Staging: GLOBAL_LOAD_ASYNC_TO_LDS_B128 into double-buffered LDS; one
// s_wait_asynccnt + one barrier per K-chunk, async copy overlapped with WMMA.
#define BM 64
#define BN 64
#define BK 16
#define LDP 20   // padded LDS row (floats): 16B-aligned fills, conflict-free frag reads

__device__ __forceinline__ void async_b128_to_lds(unsigned lds_byte_addr,
                                                  const float* gsrc) {
    // GV mode: per-lane 64-bit global address, per-lane 32-bit LDS address.
    asm volatile("global_load_async_to_lds_b128 %0, %1, off"
                 :: "v"(lds_byte_addr), "v"(gsrc) : "memory");
}
__device__ __forceinline__ void wait_async0() {
    asm volatile("s_wait_asynccnt 0" ::: "memory");
}

__global__ __launch_bounds__(256) void snn_input_gemm(
    const float* __restrict__ A,   // x  [M,K]
    const float* __restrict__ Bm,  // W  [N,K]
    float* __restrict__ C,         // i  [M,N]  (l1_seq region of d_out)
    int M, int N, int K)
{
    __shared__ float As[2][BM * LDP];
    __shared__ float Bs[2][BN * LDP];

    const int tid  = threadIdx.x;
    const int lane = tid & 31;
    const int wave = tid >> 5;          // 0..7
    const int wr   = wave & 3;          // 16-row group
    const int wc   = wave >> 2;         // 0/1 -> n offset 0/32
    const int m0   = blockIdx.x * BM;
    const int n0   = blockIdx.y * BN;

    const int lrow = tid >> 2;          // 0..63
    const int lcol = (tid & 3) << 2;    // 0,4,8,12

    // flat->LDS: byte offset lives in addr[31:0] of the shared-mem aperture
    const unsigned ldsA0 = (unsigned)(uintptr_t)&As[0][lrow * LDP + lcol];
    const unsigned ldsA1 = (unsigned)(uintptr_t)&As[1][lrow * LDP + lcol];
    const unsigned ldsB0 = (unsigned)(uintptr_t)&Bs[0][lrow * LDP + lcol];
    const unsigned ldsB1 = (unsigned)(uintptr_t)&Bs[1][lrow * LDP + lcol];

    const float* gA = A  + (size_t)(m0 + lrow) * K + lcol;
    const float* gB = Bm + (size_t)(n0 + lrow) * K + lcol;

    v8f c0 = {}; v8f c1 = {};

    const int mloc = (lane & 15);       // M/N index within 16
    const int koff = (lane >> 4) << 1;  // lanes 16..31 handle K+2,K+3

    // preload chunk 0 into buffer 0
    async_b128_to_lds(ldsA0, gA);
    async_b128_to_lds(ldsB0, gB);
    wait_async0();
    __syncthreads();

    for (int k0 = 0; k0 < K; k0 += BK) {
        const int p = (k0 / BK) & 1;
        if (k0 + BK < K) {  // prefetch next chunk into the other buffer
            async_b128_to_lds(p ? ldsA0 : ldsA1, gA + k0 + BK);
            async_b128_to_lds(p ? ldsB0 : ldsB1, gB + k0 + BK);
        }
        const float* __restrict__ Ap = As[p];
        const float* __restrict__ Bp = Bs[p];
#pragma unroll
        for (int kk = 0; kk < BK; kk += 4) {
            const int kA = kk + koff;
            // A frag 16x4: lane m holds (K,K+1) [lanes 0-15] / (K+2,K+3) [16-31]
            v2f a  = *(const v2f*)(&Ap[(wr * 16 + mloc) * LDP + kA]);
            // B frag 4x16: lane n holds same K pairing (row-of-B across lanes)
            v2f b0 = *(const v2f*)(&Bp[(wc * 32 +      mloc) * LDP + kA]);
            v2f b1 = *(const v2f*)(&Bp[(wc * 32 + 16 + mloc) * LDP + kA]);
            c0 = __builtin_amdgcn_wmma_f32_16x16x4_f32(false, a, false, b0,
                                                       (short)0, c0, false, false);
            c1 = __builtin_amdgcn_wmma_f32_16x16x4_f32(false, a, false, b1,
                                                       (short)0, c1, false, false);
        }
        wait_async0();       // next chunk landed in the other buffer
        __syncthreads();     // everyone done reading buffer p
    }

    // D layout: lane<16 -> rows r, N=lane; lane>=16 -> rows r+8, N=lane-16
    const int mo = m0 + wr * 16 + ((lane >> 4) << 3);
    const int no = n0 + wc * 32 + (lane & 15);
#pragma unroll
    for (int r = 0; r < 8; ++r) {
        C[(size_t)(mo + r) * N + no]      = c0[r];
        C[(size_t)(mo + r) * N + no + 16] = c1[r];
    }
}

// ------------------------------ Scan ---------------------------------------
// One workgroup per batch (1024 thr = 32 waves). State in LDS (~33KB of 320KB).
// Lazy-fused mem update: pass at step t reads stored m (= m_{t-2}), applies
// pending LR*t2_{t-1}⊗t1_{t-1}, uses result (= m_{t-1}) for the matvec, and
// writes it back. Final pending applied after the loop.
__global__ __launch_bounds__(1024) void snn_scan(
    float* __restrict__ memw,        // [B,O,H] working/final mem (d_out region 0)
    float* __restrict__ l1seq,       // [B,T,H] holds i on entry, spikes on exit
    float* __restrict__ l2seq,       // [B,T,O]
    const int* __restrict__ trainp)
{
    __shared__ float v1[Hdim];
    __shared__ float t1[2][Hdim];
    __shared__ float l1s[Hdim];
    __shared__ float v2[Odim];
    __shared__ float t2[2][Odim];

    const int b    = blockIdx.x;
    const int tid  = threadIdx.x;
    const int lane = tid & 31;
    const int wave = tid >> 5;       // 0..31
    const bool train = (*trainp) != 0;

    for (int h = tid; h < Hdim; h += 1024) {
        v1[h] = 0.f; t1[0][h] = 0.f; t1[1][h] = 0.f;
    }
    if (tid < Odim) { v2[tid] = 0.f; t2[0][tid] = 0.f; t2[1][tid] = 0.f; }
    __syncthreads();

    float* mb  = memw + (size_t)b * Odim * Hdim;
    float* l1b = l1seq + (size_t)b * Tdim * Hdim;
    float* l2b = l2seq + (size_t)b * Tdim * Odim;

    for (int t = 0; t < Tdim; ++t) {
        const int p = t & 1;
        const float* t1p = t1[p];
        float*       t1c = t1[p ^ 1];

        // ---- phase 1: LIF layer 1 + trace, spikes written over i in place
        float* it = l1b + (size_t)t * Hdim;
        for (int h = tid; h < Hdim; h += 1024) {
            float v = ALPHA_C * v1[h] + it[h];
            float s = (v >= THR_C) ? 1.f : 0.f;
            v -= s;                                  // soft reset (THR=1)
            v1[h]  = v;
            l1s[h] = s;
            t1c[h] = DECAY_C * t1p[h] + ONEMD_C * s;
            it[h]  = s;                              // emit l1_seq
        }
        __syncthreads();

        // prefetch next step's input currents while we stream mem
        if (t + 1 < Tdim)
            __builtin_prefetch(it + Hdim + tid * 2, 0, 3);

        // ---- phase 2: fused matvec + lazy Hebbian update, one wave per o-row
        for (int o = wave; o < Odim; o += 32) {
            const float c2 = train ? (LR_C * t2[p][o]) : 0.f;
            float* mrow = mb + (size_t)o * Hdim;
            float acc = 0.f;
            for (int h = lane * 4; h < Hdim; h += 128) {
                float4 m4        = *(float4*)(mrow + h);
                const float4 tp4 = *(const float4*)(t1p + h);
                const float4 s4  = *(const float4*)(l1s + h);
                m4.x += c2 * tp4.x; m4.y += c2 * tp4.y;
                m4.z += c2 * tp4.z; m4.w += c2 * tp4.w;
                acc += m4.x * s4.x + m4.y * s4.y + m4.z * s4.z + m4.w * s4.w;
                if (train) *(float4*)(mrow + h) = m4;
            }
            for (int off = 16; off > 0; off >>= 1)
                acc += __shfl_xor(acc, off, 32);
            if (lane == 0) {
                const float l1v = 0.2f * acc;
                float v = ALPHA_C * v2[o] + l1v;
                float s = (v >= THR_C) ? 1.f : 0.f;
                v -= s;
                v2[o] = v;
                t2[p ^ 1][o] = DECAY_C * t2[p][o] + ONEMD_C * s;
                l2b[(size_t)t * Odim + o] = s;
            }
        }
        __syncthreads();
    }

    // ---- apply last step's pending rank-1 update (T even -> buffers [0])
    if (train) {
        const int pf = Tdim & 1;
        for (int idx = tid; idx < Odim * Hdim; idx += 1024) {
            const int o = idx >> 11;           // H = 2048
            const int h = idx & (Hdim - 1);
            mb[idx] += LR_C * t2[pf][o] * t1[pf][h];
        }
    }
}

// ------------------------------ Launch -------------------------------------
extern "C" void kernel_launch(void* const* d_in, const int* in_sizes, int n_in,
                              void* d_out, int out_size, void* d_ws, size_t ws_size,
                              hipStream_t stream) {
    const float* x    = (const float*)d_in[0];   // [B,T,I]
    const float* W    = (const float*)d_in[1];   // [H,I]
    const float* mem  = (const float*)d_in[2];   // [B,O,H]
    const int*   trn  = (const int*)d_in[3];

    float* out   = (float*)d_out;
    float* memw  = out;                                          // [B,O,H]
    float* l1seq = out + (size_t)Bdim * Odim * Hdim;             // [B,T,H]
    float* l2seq = l1seq + (size_t)Bdim * Tdim * Hdim;           // [B,T,O]

    // working mem starts from the input (graph-capture-safe async d2d copy)
    hipMemcpyAsync(memw, mem, (size_t)Bdim * Odim * Hdim * sizeof(float),
                   hipMemcpyDeviceToDevice, stream);

    // i = x @ W^T  -> written directly into l1_seq region
    dim3 g((Bdim * Tdim) / BM, Hdim / BN);      // 500 x 32
    snn_input_gemm<<<g, 256, 0, stream>>>(x, W, l1seq,
                                          Bdim * Tdim, Hdim, Idim);

    // sequential LIF/Hebbian scan, one workgroup per batch
    snn_scan<<<Bdim, 1024, 0, stream>>>(memw, l1seq, l2seq, trn);
}